// H2GCNWGTL_77068893159672
// MI455X (gfx1250) — compile-verified
//
#include <hip/hip_runtime.h>
#include <hip/hip_bf16.h>

// ---------------------------------------------------------------------------
// H2GCN-with-global-topo forward, MI455X (gfx1250, wave32).
// Everything is HBM-bandwidth bound (~1.9 GB traffic); compute in f32 so the
// result matches the f32 reference. Matrix work uses V_WMMA_F32_16X16X4_F32.
// ---------------------------------------------------------------------------

#define NNODES 10000
#define IN_CH  256
#define HCH    16
#define NCLASS 10
#define BN_EPS 1e-5f

typedef __attribute__((ext_vector_type(2))) float v2f;
typedef __attribute__((ext_vector_type(8))) float v8f;

// ---------------------------------------------------------------------------
// Kernel 1: h = relu(x @ feat_W + feat_b)   [10000,256]x[256,16] -> [10000,16]
// One wave32 per 16-row tile; K=256 -> 64 v_wmma_f32_16x16x4_f32 per wave.
// feat_W (16 KB) staged in LDS, shared by the block's 8 waves.
// ---------------------------------------------------------------------------
__global__ __launch_bounds__(256) void feat_embed_kernel(
    const float* __restrict__ x, const float* __restrict__ W,
    const float* __restrict__ bias, float* __restrict__ h, int n_tiles)
{
    __shared__ float ldsW[IN_CH * HCH];            // 256x16 f32 = 16 KB
    for (int i = threadIdx.x; i < IN_CH * HCH; i += 256) ldsW[i] = W[i];
    __syncthreads();

    const int wave = threadIdx.x >> 5;
    const int lane = threadIdx.x & 31;
    const int tile = blockIdx.x * 8 + wave;
    if (tile >= n_tiles) return;                   // uniform per wave -> EXEC all-1s inside

    const int half = lane >> 4;                    // 0: lanes 0-15, 1: lanes 16-31
    const int l15  = lane & 15;
    const int koff = half * 2;                     // ISA A-frag: VGPR0/1 = K{0,1} | K{2,3}
    const float* xrow = x + (size_t)(tile * 16 + l15) * IN_CH;

    v8f acc = {};
    #pragma unroll 8
    for (int k = 0; k < IN_CH; k += 4) {
        v2f a, b;
        a.x = xrow[k + koff];                      // A: 16x4 f32, row = l15
        a.y = xrow[k + koff + 1];
        b.x = ldsW[(k + koff)     * HCH + l15];    // B: 4x16 f32, col = l15
        b.y = ldsW[(k + koff + 1) * HCH + l15];
        acc = __builtin_amdgcn_wmma_f32_16x16x4_f32(
                  false, a, false, b, (short)0, acc, false, false);
    }

    const float bv = bias[l15];
    float* hb = h + (size_t)tile * 16 * HCH;
    #pragma unroll
    for (int i = 0; i < 8; ++i) {                  // C/D layout: VGPR i -> M=i (+8 for hi half)
        const int m = half * 8 + i;
        float v = acc[i] + bv;
        hb[m * HCH + l15] = v > 0.f ? v : 0.f;     // fused ReLU
    }
}

// ---------------------------------------------------------------------------
// Kernel 2: topo = vec(2500) @ gll_W + gll_b     -> 160000 outputs
// Streams 1.6 GB of gll_W exactly once: non-temporal loads (keep the 192 MB L2
// clean) + speculative prefetch ahead of the pointer (global_prefetch_b8;
// OOB speculative prefetches are silently dropped per ISA 7.3).
// ---------------------------------------------------------------------------
__global__ __launch_bounds__(256) void topo_gemv_kernel(
    const float* __restrict__ v, const float* __restrict__ W,
    const float* __restrict__ bias, float* __restrict__ topo)
{
    __shared__ float lv[2500];
    for (int i = threadIdx.x; i < 2500; i += 256) lv[i] = v[i];
    __syncthreads();

    const int j = blockIdx.x * 256 + threadIdx.x;  // 625*256 == 160000 exactly
    const float* p = W + j;
    const size_t ldw = 160000;
    float acc = bias[j];
    for (int k = 0; k < 2500; k += 4) {            // 2500 = 4*625, no remainder
        __builtin_prefetch(p + (size_t)(k + 8) * ldw, 0, 0);
        float w0 = __builtin_nontemporal_load(p + (size_t)(k + 0) * ldw);
        float w1 = __builtin_nontemporal_load(p + (size_t)(k + 1) * ldw);
        float w2 = __builtin_nontemporal_load(p + (size_t)(k + 2) * ldw);
        float w3 = __builtin_nontemporal_load(p + (size_t)(k + 3) * ldw);
        acc = fmaf(lv[k + 0], w0, acc);
        acc = fmaf(lv[k + 1], w1, acc);
        acc = fmaf(lv[k + 2], w2, acc);
        acc = fmaf(lv[k + 3], w3, acc);
    }
    topo[j] = acc;                                 // flat == [N,H] row-major
}

// ---------------------------------------------------------------------------
// Zero a float range (c1/c2 accumulators before atomic scatter).
// ---------------------------------------------------------------------------
__global__ void zero_kernel(float* __restrict__ p, int n)
{
    int i = blockIdx.x * 256 + threadIdx.x;
    if (i < n) p[i] = 0.f;
}

// ---------------------------------------------------------------------------
// SpMM scatter: for each edge e, dst[row[e], colOff:colOff+INCH] +=
//               val[e] * src[col[e], 0:INCH]  via global_atomic_add_f32.
// ---------------------------------------------------------------------------
template <int INCH>
__global__ __launch_bounds__(256) void spmm_scatter_kernel(
    const int* __restrict__ rows, const int* __restrict__ cols,
    const float* __restrict__ vals, int nE,
    const float* __restrict__ src, float* __restrict__ dst,
    int dstStride, int colOff)
{
    int e = blockIdx.x * 256 + threadIdx.x;
    if (e >= nE) return;
    const int   r = rows[e];
    const int   c = cols[e];
    const float v = vals[e];
    const float* s = src + (size_t)c * INCH;
    float*       d = dst + (size_t)r * dstStride + colOff;
    #pragma unroll
    for (int ch = 0; ch < INCH; ch += 4) {
        float4 f = *reinterpret_cast<const float4*>(s + ch);
        atomicAdd(d + ch + 0, v * f.x);
        atomicAdd(d + ch + 1, v * f.y);
        atomicAdd(d + ch + 2, v * f.z);
        atomicAdd(d + ch + 3, v * f.w);
    }
}

// ---------------------------------------------------------------------------
// BatchNorm (eval) over c1 [N, 32], channel = i & 31.
// ---------------------------------------------------------------------------
__global__ void bn_kernel(float* __restrict__ c1,
                          const float* __restrict__ gamma,
                          const float* __restrict__ beta,
                          const float* __restrict__ mean,
                          const float* __restrict__ var, int n)
{
    int i = blockIdx.x * 256 + threadIdx.x;
    if (i >= n) return;
    int ch = i & 31;
    float inv = rsqrtf(var[ch] + BN_EPS);
    c1[i] = (c1[i] - mean[ch]) * inv * gamma[ch] + beta[ch];
}

// ---------------------------------------------------------------------------
// Final: out_feat = [xs0 | c1 | c2 | topo] (128), logits = out_feat @ fin_W + b,
// log_softmax. One thread per node (1280 FMA — negligible); fin_W in LDS.
// ---------------------------------------------------------------------------
__global__ __launch_bounds__(256) void final_kernel(
    const float* __restrict__ h,  const float* __restrict__ c1,
    const float* __restrict__ c2, const float* __restrict__ topo,
    const float* __restrict__ W,  const float* __restrict__ bias,
    float* __restrict__ out)
{
    __shared__ float lw[128 * NCLASS];             // 5 KB
    __shared__ float lb[NCLASS];
    for (int i = threadIdx.x; i < 128 * NCLASS; i += 256) lw[i] = W[i];
    if (threadIdx.x < NCLASS) lb[threadIdx.x] = bias[threadIdx.x];
    __syncthreads();

    const int node = blockIdx.x * 256 + threadIdx.x;
    if (node >= NNODES) return;

    float acc[NCLASS];
    #pragma unroll
    for (int c = 0; c < NCLASS; ++c) acc[c] = lb[c];

    const float* ph = h    + (size_t)node * 16;
    const float* p1 = c1   + (size_t)node * 32;
    const float* p2 = c2   + (size_t)node * 64;
    const float* pt = topo + (size_t)node * 16;

    for (int k = 0; k < 16; ++k) {
        float fv = ph[k];
        #pragma unroll
        for (int c = 0; c < NCLASS; ++c) acc[c] = fmaf(fv, lw[k * NCLASS + c], acc[c]);
    }
    for (int k = 0; k < 32; ++k) {
        float fv = p1[k];
        #pragma unroll
        for (int c = 0; c < NCLASS; ++c) acc[c] = fmaf(fv, lw[(16 + k) * NCLASS + c], acc[c]);
    }
    for (int k = 0; k < 64; ++k) {
        float fv = p2[k];
        #pragma unroll
        for (int c = 0; c < NCLASS; ++c) acc[c] = fmaf(fv, lw[(48 + k) * NCLASS + c], acc[c]);
    }
    for (int k = 0; k < 16; ++k) {
        float fv = pt[k];
        #pragma unroll
        for (int c = 0; c < NCLASS; ++c) acc[c] = fmaf(fv, lw[(112 + k) * NCLASS + c], acc[c]);
    }

    float m = acc[0];
    #pragma unroll
    for (int c = 1; c < NCLASS; ++c) m = fmaxf(m, acc[c]);
    float s = 0.f;
    #pragma unroll
    for (int c = 0; c < NCLASS; ++c) s += expf(acc[c] - m);
    float lse = logf(s);
    float* o = out + (size_t)node * NCLASS;
    #pragma unroll
    for (int c = 0; c < NCLASS; ++c) o[c] = acc[c] - m - lse;
}

// ---------------------------------------------------------------------------
// Launcher
// ---------------------------------------------------------------------------
extern "C" void kernel_launch(void* const* d_in, const int* in_sizes, int n_in,
                              void* d_out, int out_size, void* d_ws, size_t ws_size,
                              hipStream_t stream)
{
    const float* x     = (const float*)d_in[0];
    const float* gwc   = (const float*)d_in[1];     // 2500 floats (50x50)
    const int*   a1r   = (const int*)  d_in[2];
    const int*   a1c   = (const int*)  d_in[3];
    const float* a1v   = (const float*)d_in[4];
    const int*   a2r   = (const int*)  d_in[5];
    const int*   a2c   = (const int*)  d_in[6];
    const float* a2v   = (const float*)d_in[7];
    const float* featW = (const float*)d_in[8];
    const float* featB = (const float*)d_in[9];
    const float* gllW  = (const float*)d_in[10];
    const float* gllB  = (const float*)d_in[11];
    const float* gamma = (const float*)d_in[12];
    const float* beta  = (const float*)d_in[13];
    const float* mean  = (const float*)d_in[14];
    const float* var   = (const float*)d_in[15];
    const float* finW  = (const float*)d_in[16];
    const float* finB  = (const float*)d_in[17];
    float* out = (float*)d_out;

    const int E1 = in_sizes[2];
    const int E2 = in_sizes[5];

    // workspace layout (floats): h | topo | c1 | c2
    float* ws   = (float*)d_ws;
    float* h    = ws;                               // N*16  = 160000
    float* topo = ws + 160000;                      // N*16  = 160000
    float* c1   = ws + 320000;                      // N*32  = 320000
    float* c2   = ws + 640000;                      // N*64  = 640000

    const int n_tiles = NNODES / 16;                // 625

    // 1) feature embed (WMMA f32) — independent of topo stream order
    feat_embed_kernel<<<(n_tiles + 7) / 8, 256, 0, stream>>>(x, featW, featB, h, n_tiles);

    // 2) global topo GEMV (the 1.6 GB stream)
    topo_gemv_kernel<<<160000 / 256, 256, 0, stream>>>(gwc, gllW, gllB, topo);

    // 3) zero c1+c2 accumulators (960000 floats)
    zero_kernel<<<(960000 + 255) / 256, 256, 0, stream>>>(c1, 960000);

    // 4) conv0: 1-hop and 2-hop scatter into c1 [N,32]
    spmm_scatter_kernel<16><<<(E1 + 255) / 256, 256, 0, stream>>>(a1r, a1c, a1v, E1, h, c1, 32, 0);
    spmm_scatter_kernel<16><<<(E2 + 255) / 256, 256, 0, stream>>>(a2r, a2c, a2v, E2, h, c1, 32, 16);

    // 5) BatchNorm on c1
    bn_kernel<<<(NNODES * 32 + 255) / 256, 256, 0, stream>>>(c1, gamma, beta, mean, var, NNODES * 32);

    // 6) conv1: scatter into c2 [N,64]
    spmm_scatter_kernel<32><<<(E1 + 255) / 256, 256, 0, stream>>>(a1r, a1c, a1v, E1, c1, c2, 64, 0);
    spmm_scatter_kernel<32><<<(E2 + 255) / 256, 256, 0, stream>>>(a2r, a2c, a2v, E2, c1, c2, 64, 32);

    // 7) JK-cat + classifier + log_softmax
    final_kernel<<<(NNODES + 255) / 256, 256, 0, stream>>>(h, c1, c2, topo, finW, finB, out);
}